// GCN_model_33432025432092
// MI455X (gfx1250) — compile-verified
//
#include <hip/hip_runtime.h>
#include <hip/hip_bf16.h>
#include <math.h>

typedef __attribute__((ext_vector_type(2))) float v2f;
typedef __attribute__((ext_vector_type(8))) float v8f;

#define THREADS 256

static inline int cdiv_h(int a, int b) { return (a + b - 1) / b; }

// ---------------------------------------------------------------- small prep
__global__ void k_deg_init(float* deg, int N) {
  int i = blockIdx.x * blockDim.x + threadIdx.x;
  if (i < N) deg[i] = 1.0f;  // self-loop contributes 1 to every node's in-degree
}

__global__ void k_deg_scatter(const int* __restrict__ dst, float* deg, int E) {
  int e = blockIdx.x * blockDim.x + threadIdx.x;
  if (e < E) unsafeAtomicAdd(&deg[dst[e]], 1.0f);
}

__global__ void k_dinv(const float* __restrict__ deg, float* __restrict__ dinv,
                       float* __restrict__ selfnorm, int N) {
  int i = blockIdx.x * blockDim.x + threadIdx.x;
  if (i < N) {
    float r = rsqrtf(deg[i]);  // deg >= 1 always (self-loops)
    dinv[i] = r;
    selfnorm[i] = r * r;       // norm of the self-loop edge i->i
  }
}

__global__ void k_norm(const int* __restrict__ src, const int* __restrict__ dst,
                       const float* __restrict__ dinv, float* __restrict__ norm, int E) {
  int e = blockIdx.x * blockDim.x + threadIdx.x;
  if (e < E) norm[e] = dinv[src[e]] * dinv[dst[e]];
}

// Pad W[K x ncols] -> Wp[Kpad x 16] with zeros (rows >= K, cols >= ncols).
__global__ void k_prep_w(const float* __restrict__ W, float* __restrict__ Wp,
                         int K, int ncols, int Kpad) {
  int i = blockIdx.x * blockDim.x + threadIdx.x;
  if (i < Kpad * 16) {
    int k = i >> 4, c = i & 15;
    Wp[i] = (k < K && c < ncols) ? W[k * ncols + c] : 0.0f;
  }
}

// ------------------------------------------------- WMMA fp32 node transform
// One wave per 16-row M tile:  t[tile] = relu?(in[tile]) @ Wp   (KPAD x 16, fp32)
// Writes agg = bias + t*selfnorm directly (self-loop + bias fused; no init pass,
// each (row,col) is owned by exactly one lane).
//
// V_WMMA_F32_16X16X4_F32 layouts (ISA 7.12.2):
//   A 16x4 : lanes 0-15  -> row M=lane,    v0=K0, v1=K1
//            lanes 16-31 -> row M=lane-16, v0=K2, v1=K3
//   B 4x16 : mirrored (v0/v1 hold rows K0/K1 for lanes 0-15, K2/K3 for lanes 16-31)
//   C/D    : VGPR r = (M=r, N=lane) for lanes 0-15, (M=r+8, N=lane-16) for 16-31
template <int KPAD, int RELU>
__global__ void __launch_bounds__(THREADS)
k_transform(const float* __restrict__ in, int in_stride,
            const float* __restrict__ Wp,        // KPAD x 16, padded
            const float* __restrict__ selfnorm,  // [N]
            const float* __restrict__ bias,      // [ncols]
            int ncols,
            float* __restrict__ t,               // [N x 16] out
            float* __restrict__ agg,             // [N x 16] out = bias + t*selfnorm
            int N) {
  int wave = threadIdx.x >> 5;
  int lane = threadIdx.x & 31;
  int ntiles = (N + 15) >> 4;
  int tile = blockIdx.x * (blockDim.x >> 5) + wave;
  if (tile >= ntiles) return;          // wave-uniform: EXEC stays all-ones for WMMA

  int m0 = tile << 4;
  int half = lane >> 4;                // 0 -> K offsets {0,1}; 1 -> {2,3}
  int l15 = lane & 15;
  int arow = m0 + l15;
  if (arow > N - 1) arow = N - 1;      // clamp (N % 16 == 0 in practice)
  const float* ain = in + (size_t)arow * in_stride + 2 * half;

  v8f c = {};
#pragma unroll
  for (int k = 0; k < KPAD; k += 4) {
    float a0 = ain[k];
    float a1 = ain[k + 1];
    if (RELU) { a0 = fmaxf(a0, 0.0f); a1 = fmaxf(a1, 0.0f); }
    v2f a; a.x = a0; a.y = a1;
    v2f b;
    b.x = Wp[(k + 2 * half) * 16 + l15];
    b.y = Wp[(k + 2 * half + 1) * 16 + l15];
    c = __builtin_amdgcn_wmma_f32_16x16x4_f32(false, a, false, b,
                                              (short)0, c, false, false);
  }

  float bval = (l15 < ncols) ? bias[l15] : 0.0f;
#pragma unroll
  for (int r = 0; r < 8; ++r) {
    int row = m0 + r + 8 * half;
    if (row < N) {
      size_t o = (size_t)row * 16 + l15;
      float v = c[r];
      t[o] = v;
      agg[o] = bval + v * selfnorm[row];  // bias + self-loop, pre-atomics
    }
  }
}

// --------------------------------------------------------------- edge scatter
template <int NCOLS>
__global__ void k_scatter(const int* __restrict__ src, const int* __restrict__ dst,
                          const float* __restrict__ norm, const float* __restrict__ t,
                          float* __restrict__ agg, int E) {
  int e = blockIdx.x * blockDim.x + threadIdx.x;
  if (e >= E) return;
  int s = src[e], d = dst[e];
  float w = norm[e];
  const float4* tp = (const float4*)(t + (size_t)s * 16);  // 64B-aligned rows
  float* ad = agg + (size_t)d * 16;
  float4 q0 = tp[0];
  unsafeAtomicAdd(&ad[0], q0.x * w);
  unsafeAtomicAdd(&ad[1], q0.y * w);
  unsafeAtomicAdd(&ad[2], q0.z * w);
  unsafeAtomicAdd(&ad[3], q0.w * w);
  float4 q1 = tp[1];
  unsafeAtomicAdd(&ad[4], q1.x * w);
  unsafeAtomicAdd(&ad[5], q1.y * w);
  if (NCOLS > 6) {
    unsafeAtomicAdd(&ad[6], q1.z * w);
    unsafeAtomicAdd(&ad[7], q1.w * w);
    float4 q2 = tp[2];
    unsafeAtomicAdd(&ad[8], q2.x * w);
    unsafeAtomicAdd(&ad[9], q2.y * w);
  }
}

// ------------------------------------------------------------------- pooling
__global__ void k_pool_zero(float* sums, float* cnts) {
  int i = blockIdx.x * blockDim.x + threadIdx.x;
  if (i < 64 * 16) sums[i] = 0.0f;
  if (i < 64) cnts[i] = 0.0f;
}

__global__ void k_pool(const float* __restrict__ h, const int* __restrict__ batch,
                       float* __restrict__ sums, float* __restrict__ cnts, int N) {
  __shared__ float ls[64 * 16];
  __shared__ float lc[64];
  int tid = threadIdx.x;
  for (int i = tid; i < 64 * 16; i += blockDim.x) ls[i] = 0.0f;
  for (int i = tid; i < 64; i += blockDim.x) lc[i] = 0.0f;
  __syncthreads();
  int i = blockIdx.x * blockDim.x + tid;
  if (i < N) {
    int g = batch[i];
    const float4* hp = (const float4*)(h + (size_t)i * 16);
    float4 q0 = hp[0], q1 = hp[1], q2 = hp[2];
    float v[10] = {q0.x, q0.y, q0.z, q0.w, q1.x, q1.y, q1.z, q1.w, q2.x, q2.y};
#pragma unroll
    for (int c = 0; c < 10; ++c)
      unsafeAtomicAdd(&ls[g * 16 + c], fmaxf(v[c], 0.0f));  // final ReLU here
    unsafeAtomicAdd(&lc[g], 1.0f);
  }
  __syncthreads();
  if (tid < 64 && lc[tid] > 0.0f) {
#pragma unroll
    for (int c = 0; c < 10; ++c)
      unsafeAtomicAdd(&sums[tid * 16 + c], ls[tid * 16 + c]);
    unsafeAtomicAdd(&cnts[tid], lc[tid]);
  }
}

__global__ void k_final(const float* __restrict__ sums, const float* __restrict__ cnts,
                        float* __restrict__ out) {
  int g = threadIdx.x;
  if (g >= 64) return;
  float cnt = fmaxf(cnts[g], 1.0f);
  float p[10];
  float m = -1e30f;
#pragma unroll
  for (int c = 0; c < 10; ++c) { p[c] = sums[g * 16 + c] / cnt; m = fmaxf(m, p[c]); }
  float s = 0.0f;
#pragma unroll
  for (int c = 0; c < 10; ++c) s += expf(p[c] - m);
  float l = logf(s);
#pragma unroll
  for (int c = 0; c < 10; ++c) out[g * 10 + c] = p[c] - m - l;
}

// ------------------------------------------------------------------ launcher
extern "C" void kernel_launch(void* const* d_in, const int* in_sizes, int n_in,
                              void* d_out, int out_size, void* d_ws, size_t ws_size,
                              hipStream_t stream) {
  const int NF = 128;
  const float* x = (const float*)d_in[0];
  const int* edge = (const int*)d_in[1];
  const int* batch = (const int*)d_in[2];
  const float* W[7] = {(const float*)d_in[3],  (const float*)d_in[5],
                       (const float*)d_in[7],  (const float*)d_in[9],
                       (const float*)d_in[11], (const float*)d_in[13],
                       (const float*)d_in[15]};
  const float* B[7] = {(const float*)d_in[4],  (const float*)d_in[6],
                       (const float*)d_in[8],  (const float*)d_in[10],
                       (const float*)d_in[12], (const float*)d_in[14],
                       (const float*)d_in[16]};

  int N = in_sizes[0] / NF;   // 100000
  int E = in_sizes[1] / 2;    // 3200000
  const int* src = edge;      // edge_index row 0
  const int* dst = edge + E;  // edge_index row 1

  // workspace carve-up (floats)
  float* w = (float*)d_ws;
  size_t off = 0;
  float* deg      = w + off; off += (size_t)N;
  float* dinv     = w + off; off += (size_t)N;
  float* selfnorm = w + off; off += (size_t)N;
  float* norm     = w + off; off += (size_t)E;
  float* t        = w + off; off += (size_t)N * 16;
  float* aggbuf[2];
  aggbuf[0]       = w + off; off += (size_t)N * 16;
  aggbuf[1]       = w + off; off += (size_t)N * 16;
  float* Wp       = w + off; off += (size_t)7 * 2048;
  float* sums     = w + off; off += 64 * 16;
  float* cnts     = w + off; off += 64;

  int Kin[7]   = {128, 6, 6, 6, 6, 6, 6};
  int Kpad[7]  = {128, 8, 8, 8, 8, 8, 8};
  int ncols[7] = {6, 6, 6, 6, 6, 6, 10};

  for (int l = 0; l < 7; ++l)
    k_prep_w<<<cdiv_h(Kpad[l] * 16, THREADS), THREADS, 0, stream>>>(
        W[l], Wp + l * 2048, Kin[l], ncols[l], Kpad[l]);

  k_deg_init<<<cdiv_h(N, THREADS), THREADS, 0, stream>>>(deg, N);
  k_deg_scatter<<<cdiv_h(E, THREADS), THREADS, 0, stream>>>(dst, deg, E);
  k_dinv<<<cdiv_h(N, THREADS), THREADS, 0, stream>>>(deg, dinv, selfnorm, N);
  k_norm<<<cdiv_h(E, THREADS), THREADS, 0, stream>>>(src, dst, dinv, norm, E);

  int ntiles = cdiv_h(N, 16);
  int wpb = THREADS / 32;  // waves per block
  int tgrid = cdiv_h(ntiles, wpb);
  int egrid = cdiv_h(E, THREADS);

  // layer 1: x (K=128, no input relu) -> aggbuf[0]
  k_transform<128, 0><<<tgrid, THREADS, 0, stream>>>(
      x, NF, Wp + 0 * 2048, selfnorm, B[0], ncols[0], t, aggbuf[0], N);
  k_scatter<6><<<egrid, THREADS, 0, stream>>>(src, dst, norm, t, aggbuf[0], E);

  // layers 2..6: hidden 6->6 (K padded to 8, relu on input)
  for (int l = 1; l < 6; ++l) {
    const float* in = aggbuf[(l + 1) & 1];
    float* agg = aggbuf[l & 1];
    k_transform<8, 1><<<tgrid, THREADS, 0, stream>>>(
        in, 16, Wp + l * 2048, selfnorm, B[l], ncols[l], t, agg, N);
    k_scatter<6><<<egrid, THREADS, 0, stream>>>(src, dst, norm, t, agg, E);
  }

  // layer 7: 6->10, output in aggbuf[0]
  k_transform<8, 1><<<tgrid, THREADS, 0, stream>>>(
      aggbuf[1], 16, Wp + 6 * 2048, selfnorm, B[6], ncols[6], t, aggbuf[0], N);
  k_scatter<10><<<egrid, THREADS, 0, stream>>>(src, dst, norm, t, aggbuf[0], E);

  // global mean pool + log_softmax
  k_pool_zero<<<cdiv_h(64 * 16, THREADS), THREADS, 0, stream>>>(sums, cnts);
  k_pool<<<cdiv_h(N, THREADS), THREADS, 0, stream>>>(aggbuf[0], batch, sums, cnts, N);
  k_final<<<1, 64, 0, stream>>>(sums, cnts, (float*)d_out);
}